// Simulator_44401371906215
// MI455X (gfx1250) — compile-verified
//
#include <hip/hip_runtime.h>

// ---------------------------------------------------------------------------
// Explicit-Euler rollout of x' = A x + B u as a parallel affine scan.
//   M = I + dt*A (3x3 const), b_i = dt*(B u_i).  x_{i+1} = M x_i + b_i.
// Phase 1: per-block forcing totals (WMMA computes b; affine LDS scan).
// Phase 2: scan of block totals with V = M^4096 -> per-block start states.
// Phase 3: re-derive per-thread start state exactly, roll 16 steps,
//          stage xsol/ys in LDS, flush with coalesced stores.
// ---------------------------------------------------------------------------

#define DT_F 900.0f
#define BT   256                 // threads per block (8 wave32)
#define SEG  16                  // serial steps per thread
#define SPB  (BT * SEG)          // 4096 steps per block

typedef __attribute__((ext_vector_type(2))) float v2f;
typedef __attribute__((ext_vector_type(8))) float v8f;

// ---------------- 3x3 helpers (row-major) ----------------
__device__ __forceinline__ void mm33(const float* A, const float* B, float* C) {
#pragma unroll
  for (int r = 0; r < 3; ++r)
#pragma unroll
    for (int c = 0; c < 3; ++c)
      C[r * 3 + c] = A[r * 3 + 0] * B[0 * 3 + c] +
                     A[r * 3 + 1] * B[1 * 3 + c] +
                     A[r * 3 + 2] * B[2 * 3 + c];
}
__device__ __forceinline__ void matsq(float* P) {
  float T[9];
  mm33(P, P, T);
#pragma unroll
  for (int i = 0; i < 9; ++i) P[i] = T[i];
}
__device__ __forceinline__ void mv3(const float* A, const float* x, float* y) {
  float t0 = A[0] * x[0] + A[1] * x[1] + A[2] * x[2];
  float t1 = A[3] * x[0] + A[4] * x[1] + A[5] * x[2];
  float t2 = A[6] * x[0] + A[7] * x[1] + A[8] * x[2];
  y[0] = t0; y[1] = t1; y[2] = t2;
}
__device__ __forceinline__ void build_M(const float* Amat, float* M) {
#pragma unroll
  for (int r = 0; r < 3; ++r)
#pragma unroll
    for (int c = 0; c < 3; ++c)
      M[r * 3 + c] = ((r == c) ? 1.0f : 0.0f) + DT_F * Amat[r * 3 + c];
}
__device__ __forceinline__ void build_W16(const float* M, float* W) {
  // W = M^16 via 4 squarings
  float T[9];
  mm33(M, M, T);          // M^2
  mm33(T, T, W);          // M^4
  mm33(W, W, T);          // M^8
  mm33(T, T, W);          // M^16
}

// ---------------- WMMA forcing fill ----------------
// Computes b_i = dt*(B u_i) for all 4096 steps of this block into sB
// (sB[step*3 + c]).  One v_wmma_f32_16x16x4_f32 tile covers 16 steps:
//   D[16x16] = Apad[16x4] x Uop[4x16],  Apad rows = state comps of dt*B,
//   Uop cols = time steps.  K=5 is split into two chained WMMAs (K 0..3, 4..7).
// D layout (32-bit C/D 16x16): lanes 0-15, VGPR v -> D[M=v][N=lane],
// so lanes 0-15 hold the full b-vector of step (tileBase+lane) in acc[0..2].
__device__ __forceinline__ void fill_b_wmma(const float* __restrict__ u,
                                            long long N, long long blockBase,
                                            const float* Bdt /*dt*B, 15*/,
                                            float* sB) {
  const int tid  = threadIdx.x;
  const int lane = tid & 31;
  const int wave = tid >> 5;
  const int m    = lane & 15;   // A-row (state comp) / U-row (step-in-tile)
  const int h    = lane >> 4;   // half: K-group select

  // A-operand (constant across tiles): dt*B zero-padded to 16x8.
  // 16x4 f32 A layout: lane l holds A[l%16][2*(l>>4) + vgpr].
  v2f a1, a2;
  a1.x = (m < 3)            ? Bdt[m * 5 + 2 * h]     : 0.0f;  // k = 2h
  a1.y = (m < 3)            ? Bdt[m * 5 + 2 * h + 1] : 0.0f;  // k = 2h+1
  a2.x = (m < 3 && h == 0)  ? Bdt[m * 5 + 4]         : 0.0f;  // k = 4 (real)
  a2.y = 0.0f;                                                 // k = 5/7 pad

  const long long waveBase = blockBase + (long long)wave * (32 * SEG);
#pragma unroll 1
  for (int t = 0; t < 32; ++t) {
    long long row = waveBase + (long long)t * SEG + m;   // time step index
    // B-operand 4x16: lane l, vgpr j -> Uop[k = 2h + j][n = l%16] = u[row][k]
    v2f b1; b1.x = 0.0f; b1.y = 0.0f;
    v2f b2; b2.x = 0.0f; b2.y = 0.0f;
    if (row < N) {
      const float* up = u + row * 5;
      b1.x = up[2 * h];
      b1.y = up[2 * h + 1];
      if (h == 0) b2.x = up[4];
      __builtin_prefetch((const void*)(up + 5 * 2048), 0, 3);  // stream hint
    }
    v8f acc = {};
    acc = __builtin_amdgcn_wmma_f32_16x16x4_f32(false, a1, false, b1,
                                                (short)0, acc, false, false);
    acc = __builtin_amdgcn_wmma_f32_16x16x4_f32(false, a2, false, b2,
                                                (short)0, acc, false, false);
    if (h == 0) {
      // step-in-block = (wave*32 + t)*16 + lane  (owned by thread wave*32+t)
      float* d = &sB[(((long long)wave * 32 + t) * SEG + m) * 3];
      d[0] = acc[0]; d[1] = acc[1]; d[2] = acc[2];
    }
  }
}

// ---------------- per-thread segment forcing + block affine scan ----------
// On exit: F = inclusive forcing prefix of this thread's run; sF holds all
// inclusive prefixes (sF[t*3..] = forcing of steps [0 .. 16*(t+1)) in block).
__device__ __forceinline__ void seg_and_scan(const float* sB, const float* M,
                                             const float* W, float* sF,
                                             float F[3]) {
  const int tid = threadIdx.x;
  F[0] = 0.0f; F[1] = 0.0f; F[2] = 0.0f;
  const float* bp = &sB[tid * SEG * 3];
#pragma unroll
  for (int j = 0; j < SEG; ++j) {
    float t0 = M[0] * F[0] + M[1] * F[1] + M[2] * F[2] + bp[j * 3 + 0];
    float t1 = M[3] * F[0] + M[4] * F[1] + M[5] * F[2] + bp[j * 3 + 1];
    float t2 = M[6] * F[0] + M[7] * F[1] + M[8] * F[2] + bp[j * 3 + 2];
    F[0] = t0; F[1] = t1; F[2] = t2;
  }
  sF[tid * 3 + 0] = F[0]; sF[tid * 3 + 1] = F[1]; sF[tid * 3 + 2] = F[2];
  __syncthreads();
  float P[9];
#pragma unroll
  for (int i = 0; i < 9; ++i) P[i] = W[i];       // P = M^(16*2^k)
  for (int s = 1; s < BT; s <<= 1) {
    float l0 = 0.0f, l1 = 0.0f, l2 = 0.0f;
    if (tid >= s) {
      l0 = sF[(tid - s) * 3 + 0];
      l1 = sF[(tid - s) * 3 + 1];
      l2 = sF[(tid - s) * 3 + 2];
    }
    __syncthreads();
    if (tid >= s) {
      F[0] = P[0] * l0 + P[1] * l1 + P[2] * l2 + F[0];
      F[1] = P[3] * l0 + P[4] * l1 + P[5] * l2 + F[1];
      F[2] = P[6] * l0 + P[7] * l1 + P[8] * l2 + F[2];
    }
    sF[tid * 3 + 0] = F[0]; sF[tid * 3 + 1] = F[1]; sF[tid * 3 + 2] = F[2];
    __syncthreads();
    matsq(P);
  }
}

// ---------------- Phase 1: block forcing totals ----------------
__global__ __launch_bounds__(BT) void k_phase1(const float* __restrict__ u,
                                               const float* __restrict__ Amat,
                                               const float* __restrict__ Bmat,
                                               long long N,
                                               float* __restrict__ blkTot) {
  __shared__ __align__(16) float sB[SPB * 3];   // 48 KB forcing buffer
  __shared__ float sF[BT * 3];
  float M[9], W[9], Bdt[15];
  build_M(Amat, M);
  build_W16(M, W);
#pragma unroll
  for (int i = 0; i < 15; ++i) Bdt[i] = DT_F * Bmat[i];

  const long long blockBase = (long long)blockIdx.x * SPB;
  fill_b_wmma(u, N, blockBase, Bdt, sB);
  __syncthreads();

  float F[3];
  seg_and_scan(sB, M, W, sF, F);
  if (threadIdx.x == BT - 1) {
    blkTot[blockIdx.x * 3 + 0] = F[0];
    blkTot[blockIdx.x * 3 + 1] = F[1];
    blkTot[blockIdx.x * 3 + 2] = F[2];
  }
}

// ---------------- Phase 2: scan of block totals -> block start states ------
__global__ __launch_bounds__(1024) void k_phase2(const float* __restrict__ xinit,
                                                 const float* __restrict__ Amat,
                                                 int NB,
                                                 const float* __restrict__ blkTot,
                                                 float* __restrict__ blkStart) {
  __shared__ float sF[1024 * 3];
  const int tid = threadIdx.x;
  float M[9], V[9];
  build_M(Amat, M);
#pragma unroll
  for (int i = 0; i < 9; ++i) V[i] = M[i];
  for (int k = 0; k < 12; ++k) matsq(V);         // V = M^4096

  float F[3] = {0.0f, 0.0f, 0.0f};
  if (tid < NB) {
    F[0] = blkTot[tid * 3 + 0];
    F[1] = blkTot[tid * 3 + 1];
    F[2] = blkTot[tid * 3 + 2];
  }
  sF[tid * 3 + 0] = F[0]; sF[tid * 3 + 1] = F[1]; sF[tid * 3 + 2] = F[2];
  __syncthreads();
  float P[9];
#pragma unroll
  for (int i = 0; i < 9; ++i) P[i] = V[i];
  for (int s = 1; s < 1024; s <<= 1) {
    float l0 = 0.0f, l1 = 0.0f, l2 = 0.0f;
    if (tid >= s) {
      l0 = sF[(tid - s) * 3 + 0];
      l1 = sF[(tid - s) * 3 + 1];
      l2 = sF[(tid - s) * 3 + 2];
    }
    __syncthreads();
    if (tid >= s) {
      F[0] = P[0] * l0 + P[1] * l1 + P[2] * l2 + F[0];
      F[1] = P[3] * l0 + P[4] * l1 + P[5] * l2 + F[1];
      F[2] = P[6] * l0 + P[7] * l1 + P[8] * l2 + F[2];
    }
    sF[tid * 3 + 0] = F[0]; sF[tid * 3 + 1] = F[1]; sF[tid * 3 + 2] = F[2];
    __syncthreads();
    matsq(P);
  }

  if (tid < NB && NB <= 1024) {
    // blkStart[b] = V^b * x0 + exclusive_prefix(b)
    float x[3] = {xinit[0], xinit[1], xinit[2]};
    float Q[9];
#pragma unroll
    for (int i = 0; i < 9; ++i) Q[i] = V[i];
    for (int k = 0; k < 10; ++k) {
      if ((tid >> k) & 1) mv3(Q, x, x);
      matsq(Q);
    }
    if (tid > 0) {
      x[0] += sF[(tid - 1) * 3 + 0];
      x[1] += sF[(tid - 1) * 3 + 1];
      x[2] += sF[(tid - 1) * 3 + 2];
    }
    blkStart[tid * 3 + 0] = x[0];
    blkStart[tid * 3 + 1] = x[1];
    blkStart[tid * 3 + 2] = x[2];
  }
  __syncthreads();
  if (NB > 1024 && tid == 0) {        // generic fallback (unused for N=4e6)
    float x[3] = {xinit[0], xinit[1], xinit[2]};
    for (int b = 0; b < NB; ++b) {
      blkStart[b * 3 + 0] = x[0];
      blkStart[b * 3 + 1] = x[1];
      blkStart[b * 3 + 2] = x[2];
      mv3(V, x, x);
      x[0] += blkTot[b * 3 + 0];
      x[1] += blkTot[b * 3 + 1];
      x[2] += blkTot[b * 3 + 2];
    }
  }
}

// ---------------- Phase 3: rollout + staged, coalesced output ----------------
__global__ __launch_bounds__(BT) void k_phase3(const float* __restrict__ u,
                                               const float* __restrict__ Amat,
                                               const float* __restrict__ Bmat,
                                               const float* __restrict__ Cmat,
                                               long long N,
                                               const float* __restrict__ blkStart,
                                               float* __restrict__ out) {
  __shared__ __align__(16) float sB[SPB * 3];
  __shared__ float sF[BT * 3];
  float M[9], W[9], Bdt[15];
  build_M(Amat, M);
  build_W16(M, W);
#pragma unroll
  for (int i = 0; i < 15; ++i) Bdt[i] = DT_F * Bmat[i];

  const long long blockBase = (long long)blockIdx.x * SPB;
  fill_b_wmma(u, N, blockBase, Bdt, sB);
  __syncthreads();

  float F[3];
  seg_and_scan(sB, M, W, sF, F);       // sF: inclusive prefixes (synced)

  const int tid = threadIdx.x;
  // x at this thread's segment start = W^tid * blkStart + exclusive_prefix
  float x[3] = {blkStart[blockIdx.x * 3 + 0],
                blkStart[blockIdx.x * 3 + 1],
                blkStart[blockIdx.x * 3 + 2]};
  float P[9];
#pragma unroll
  for (int i = 0; i < 9; ++i) P[i] = W[i];
  for (int k = 0; k < 8; ++k) {
    if ((tid >> k) & 1) mv3(P, x, x);
    matsq(P);
  }
  if (tid > 0) {
    x[0] += sF[(tid - 1) * 3 + 0];
    x[1] += sF[(tid - 1) * 3 + 1];
    x[2] += sF[(tid - 1) * 3 + 2];
  }

  // Rollout.  Each step: emit y_i (register), overwrite the consumed b_i slot
  // in LDS with the PRE-update state x_i (== xsol row i), advance x.
  const float c0 = Cmat[0], c1 = Cmat[1], c2 = Cmat[2];
  float* slot = &sB[tid * SEG * 3];
  const long long base = blockBase + (long long)tid * SEG;
  float ysv[SEG];
#pragma unroll
  for (int j = 0; j < SEG; ++j) {
    const long long i = base + j;
    if (i < N) {
      const float b0 = slot[j * 3 + 0];
      const float b1 = slot[j * 3 + 1];
      const float b2 = slot[j * 3 + 2];
      ysv[j] = c0 * x[0] + c1 * x[1] + c2 * x[2];      // y_i: pre-update x
      slot[j * 3 + 0] = x[0];                           // xsol row i (pre)
      slot[j * 3 + 1] = x[1];
      slot[j * 3 + 2] = x[2];
      const float t0 = M[0] * x[0] + M[1] * x[1] + M[2] * x[2] + b0;
      const float t1 = M[3] * x[0] + M[4] * x[1] + M[5] * x[2] + b1;
      const float t2 = M[6] * x[0] + M[7] * x[1] + M[8] * x[2] + b2;
      x[0] = t0; x[1] = t1; x[2] = t2;
      if (i == N - 1) {                                 // final xsol row N
        out[(i + 1) * 3 + 0] = t0;
        out[(i + 1) * 3 + 1] = t1;
        out[(i + 1) * 3 + 2] = t2;
      }
    } else {
      ysv[j] = 0.0f;
    }
  }
  __syncthreads();

  // Flush xsol rows [blockBase, blockBase+rows) with coalesced stores.
  // blockBase*3 floats = blockIdx*48KB -> 16B aligned.
  const long long remain = N - blockBase;
  const long long rows   = (remain < (long long)SPB) ? remain : (long long)SPB;
  float* __restrict__ dst = out + blockBase * 3;
  if (rows == SPB) {
    const float4* s4 = (const float4*)sB;
    float4* __restrict__ d4 = (float4*)dst;
#pragma unroll
    for (int q = 0; q < (SPB * 3) / (4 * BT); ++q)      // 12 float4 per thread
      d4[q * BT + tid] = s4[q * BT + tid];
  } else {
    for (long long i = tid; i < rows * 3; i += BT) dst[i] = sB[i];
  }
  __syncthreads();

  // Stage ys through the (now free) LDS buffer, flush coalesced.
#pragma unroll
  for (int j = 0; j < SEG; ++j) sB[tid * SEG + j] = ysv[j];
  __syncthreads();
  float* __restrict__ ydst = out + (N + 1) * 3 + blockBase;
  if (rows == SPB) {
#pragma unroll
    for (int q = 0; q < SPB / BT; ++q)                  // 16 dwords per thread
      ydst[q * BT + tid] = sB[q * BT + tid];
  } else {
    for (long long i = tid; i < rows; i += BT) ydst[i] = sB[i];
  }
}

// ---------------- host launcher ----------------
extern "C" void kernel_launch(void* const* d_in, const int* in_sizes, int n_in,
                              void* d_out, int out_size, void* d_ws,
                              size_t ws_size, hipStream_t stream) {
  const float* x_init = (const float*)d_in[0];
  const float* u      = (const float*)d_in[1];
  const float* Amat   = (const float*)d_in[2];
  const float* Bmat   = (const float*)d_in[3];
  const float* Cmat   = (const float*)d_in[4];
  const long long N   = (long long)in_sizes[1] / 5;
  const int NB        = (int)((N + SPB - 1) / SPB);

  float* ws       = (float*)d_ws;
  float* blkTot   = ws;                       // NB*3 floats
  float* blkStart = ws + (size_t)NB * 3;      // NB*3 floats
  float* out      = (float*)d_out;

  k_phase1<<<NB, BT, 0, stream>>>(u, Amat, Bmat, N, blkTot);
  k_phase2<<<1, 1024, 0, stream>>>(x_init, Amat, NB, blkTot, blkStart);
  k_phase3<<<NB, BT, 0, stream>>>(u, Amat, Bmat, Cmat, N, blkStart, out);
}